// ProposalLayer_6511170421104
// MI455X (gfx1250) — compile-verified
//
#include <hip/hip_runtime.h>
#include <stdint.h>

#define Hh 64
#define Ww 64
#define Aa 9
#define NANCH (Hh*Ww*Aa)      /* 36864 */
#define BATCH 8
#define PRE_NMS 2000
#define POST_NMS 300
#define IOU_T 0.7f
#define NBIN 8192             /* histogram bins on ordered-float >> 19 */
#define SORT_N 4096           /* bitonic sort capacity (candidates <= ~3200) */
#define MASK_W 63             /* ceil(2000/32) */
#define MASK_STRIDE 64
#define MASK_THREADS 1024     /* kernel-2 block size: 32 waves */

typedef unsigned int  u32x4 __attribute__((ext_vector_type(4)));
typedef int           i32x8 __attribute__((ext_vector_type(8)));
typedef int           i32x4 __attribute__((ext_vector_type(4)));

__device__ __forceinline__ unsigned ord_f32(float v) {
  unsigned u = __float_as_uint(v);
  return (u & 0x80000000u) ? ~u : (u | 0x80000000u);  // order-preserving map
}

// ---------------------------------------------------------------------------
// Kernel 1: per-batch top-2000 selection (histogram-select + bitonic sort of
// composite keys) + box decode for winners only. One 32-wave block per batch.
// ---------------------------------------------------------------------------
__global__ __launch_bounds__(1024) void topk_decode_kernel(
    const float* __restrict__ labels, const float* __restrict__ deltas,
    const float* __restrict__ base_anchors, float* __restrict__ pre_boxes)
{
  __shared__ __align__(16) char raw[SORT_N * 8];     // 32KB: hist then keys
  unsigned*            hist = (unsigned*)raw;
  unsigned long long*  keys = (unsigned long long*)raw;
  __shared__ unsigned chunksum[256];
  __shared__ unsigned s_thresh, s_count;
  __shared__ float    s_base[36];

  const int b   = blockIdx.x;
  const int tid = threadIdx.x;
  const float* lab = labels + (size_t)b * NANCH;

  if (tid < 36) s_base[tid] = base_anchors[tid];
  for (int t = tid; t < NBIN; t += 1024) hist[t] = 0;
  __syncthreads();

  // Pass 1: histogram of ordered-float high bits
  for (int n = tid; n < NANCH; n += 1024) {
    if (n + 1024 < NANCH) __builtin_prefetch(lab + n + 1024, 0, 0);
    unsigned o = ord_f32(lab[n]);
    atomicAdd(&hist[o >> 19], 1u);
  }
  __syncthreads();

  // Suffix-count to find threshold bin (largest t with count(>=t) >= 2000)
  if (tid < 256) {
    unsigned s = 0;
    for (int k = 0; k < 32; ++k) s += hist[tid * 32 + k];
    chunksum[tid] = s;
  }
  __syncthreads();
  if (tid == 0) {
    unsigned cum = 0; int c = 255;
    for (; c > 0; --c) { if (cum + chunksum[c] >= PRE_NMS) break; cum += chunksum[c]; }
    int bin = c * 32 + 31;
    for (;; --bin) { cum += hist[bin]; if (cum >= PRE_NMS || bin == c * 32) break; }
    s_thresh = (unsigned)bin;
    s_count  = 0;
  }
  __syncthreads();
  unsigned thresh = s_thresh;
  for (int t = tid; t < SORT_N; t += 1024) keys[t] = 0ull;  // pad = smallest key
  __syncthreads();

  // Pass 2: compact candidates as 64-bit keys (value desc, index asc on ties)
  for (int n = tid; n < NANCH; n += 1024) {
    if (n + 1024 < NANCH) __builtin_prefetch(lab + n + 1024, 0, 0);
    unsigned o = ord_f32(lab[n]);
    if ((o >> 19) >= thresh) {
      unsigned pos = atomicAdd(&s_count, 1u);
      if (pos < SORT_N)
        keys[pos] = ((unsigned long long)o << 32) | (unsigned)(~n);
    }
  }
  __syncthreads();

  // Bitonic sort descending, 4096 u64 keys in LDS
  for (unsigned k = 2; k <= SORT_N; k <<= 1) {
    for (unsigned j = k >> 1; j > 0; j >>= 1) {
      for (unsigned t = tid; t < SORT_N; t += 1024) {
        unsigned ixj = t ^ j;
        if (ixj > t) {
          unsigned long long a = keys[t], c2 = keys[ixj];
          if (((t & k) == 0) ? (a < c2) : (a > c2)) { keys[t] = c2; keys[ixj] = a; }
        }
      }
      __syncthreads();
    }
  }

  // Decode boxes only for the 2000 winners (score-descending order)
  for (int kk = tid; kk < PRE_NMS; kk += 1024) {
    unsigned long long key = keys[kk];
    unsigned n = ~(unsigned)(key & 0xFFFFFFFFull);
    unsigned h = n / (Ww * Aa);
    unsigned r = n - h * (Ww * Aa);
    unsigned w = r / Aa;
    unsigned a = r - w * Aa;
    float cy = ((float)h + 0.5f) * (1.0f / Hh);
    float cx = ((float)w + 0.5f) * (1.0f / Ww);
    float ay0 = fminf(fmaxf(cy + s_base[a * 4 + 0], 0.f), 1.f);
    float ax0 = fminf(fmaxf(cx + s_base[a * 4 + 1], 0.f), 1.f);
    float ay1 = fminf(fmaxf(cy + s_base[a * 4 + 2], 0.f), 1.f);
    float ax1 = fminf(fmaxf(cx + s_base[a * 4 + 3], 0.f), 1.f);
    float ah = ay1 - ay0, aw = ax1 - ax0;
    float acy = ay0 + 0.5f * ah, acx = ax0 + 0.5f * aw;
    const float* dp = deltas + ((size_t)b * NANCH + n) * 4;
    float d0 = dp[0] * 0.1f, d1 = dp[1] * 0.1f;
    float d2 = dp[2] * 0.2f, d3 = dp[3] * 0.2f;
    float bh = __expf(d2) * ah, bw = __expf(d3) * aw;
    float bcy = d0 * ah + acy, bcx = d1 * aw + acx;
    ((float4*)pre_boxes)[(size_t)b * PRE_NMS + kk] =
        make_float4(bcy - 0.5f * bh, bcx - 0.5f * bw, bcy + 0.5f * bh, bcx + 0.5f * bw);
  }
}

// ---------------------------------------------------------------------------
// Kernel 2: suppression bit-matrix. Box tile (2000 x float4 = 32KB) staged in
// LDS via one TDM tensor_load_to_lds per block (32-wave blocks amortize the
// staging 4x vs 256-thread blocks); each thread computes one (row, 32-col
// word) of the upper-triangular IoU>0.7 bit matrix.
// ---------------------------------------------------------------------------
__global__ __launch_bounds__(MASK_THREADS) void build_mask_kernel(
    const float* __restrict__ pre_boxes, unsigned* __restrict__ mask)
{
  __shared__ __align__(16) float4 sbox[PRE_NMS];     // 32KB
  const int b   = blockIdx.x;
  const int tid = threadIdx.x;

  if (threadIdx.x < 32) {  // wave 0 issues the TDM DMA
    unsigned long long ga = (unsigned long long)(uintptr_t)(pre_boxes + (size_t)b * PRE_NMS * 4);
    unsigned lds = (unsigned)(uintptr_t)&sbox[0];
    // D# group0: count=1 | lds_addr | global_addr[56:0] | type=2
    u32x4 g0 = { 1u, lds, (unsigned)ga,
                 ((unsigned)((ga >> 32) & 0x01FFFFFFull)) | (2u << 30) };
    // D# group1: data_size=4B, tensor_dim0=8000, tensor_dim1=1, tile_dim0=8000,
    //            tensor_dim0_stride=8000 (one-row tile, 32KB)
    i32x8 g1 = { (int)0x00020000, (int)(8000u << 16), (int)(1u << 16),
                 (int)(8000u << 16), 0, 8000, 0, 0 };
    i32x4 gz4 = { 0, 0, 0, 0 };
    i32x8 gz8 = { 0, 0, 0, 0, 0, 0, 0, 0 };
    __builtin_amdgcn_tensor_load_to_lds(g0, g1, gz4, gz4, gz8, 0);
    __builtin_amdgcn_s_wait_tensorcnt(0);
  }
  __syncthreads();

  int p = blockIdx.y * MASK_THREADS + tid;
  if (p >= PRE_NMS * MASK_W) return;
  int row = p / MASK_W;
  int w   = p - row * MASK_W;
  float4 bi = sbox[row];
  float  ai = (bi.z - bi.x) * (bi.w - bi.y);
  unsigned bits = 0;
  int j0 = w << 5;
  #pragma unroll 4
  for (int jj = 0; jj < 32; ++jj) {
    int j = j0 + jj;
    if (j < PRE_NMS && j > row) {
      float4 bj = sbox[j];
      float aj = (bj.z - bj.x) * (bj.w - bj.y);
      float ty = fmaxf(bi.x, bj.x), tx = fmaxf(bi.y, bj.y);
      float by = fminf(bi.z, bj.z), bx = fminf(bi.w, bj.w);
      float hh = fmaxf(by - ty, 0.f), ww2 = fmaxf(bx - tx, 0.f);
      float inter = hh * ww2;
      float iou = inter / (ai + aj - inter + 1e-8f);
      if (iou > IOU_T) bits |= (1u << jj);
    }
  }
  mask[((size_t)b * PRE_NMS + row) * MASK_STRIDE + w] = bits;
}

// ---------------------------------------------------------------------------
// Kernel 3: serial greedy scan over the bit-matrix (2-wave block per batch;
// lanes own the 63 bitmask words), then emit first <=300 kept boxes + zeros.
// Bit i of row i is never set (j>i only), so the intra-iteration OR cannot
// corrupt the kept-test for i; one barrier per step suffices.
// ---------------------------------------------------------------------------
__global__ __launch_bounds__(64) void nms_scan_kernel(
    const float* __restrict__ pre_boxes, const unsigned* __restrict__ mask,
    float* __restrict__ out)
{
  __shared__ unsigned rem[MASK_W];
  __shared__ int keep_list[POST_NMS];
  __shared__ int nkeep;
  const int b   = blockIdx.x;
  const int tid = threadIdx.x;
  if (tid < MASK_W) rem[tid] = 0;
  if (tid == 0) nkeep = 0;
  __syncthreads();
  const unsigned* mrow = mask + (size_t)b * PRE_NMS * MASK_STRIDE;
  for (int i = 0; i < PRE_NMS; ++i) {
    bool kept = ((rem[i >> 5] >> (i & 31)) & 1u) == 0u;
    if (kept) {
      if (tid < MASK_W) rem[tid] |= mrow[(size_t)i * MASK_STRIDE + tid];
      if (tid == 0) { if (nkeep < POST_NMS) keep_list[nkeep] = i; nkeep++; }
    }
    __syncthreads();
  }
  int nk = nkeep < POST_NMS ? nkeep : POST_NMS;
  for (int k = tid; k < POST_NMS; k += blockDim.x) {
    float4 v = make_float4(0.f, 0.f, 0.f, 0.f);
    if (k < nk) v = ((const float4*)pre_boxes)[(size_t)b * PRE_NMS + keep_list[k]];
    ((float4*)out)[(size_t)b * POST_NMS + k] = v;
  }
}

// ---------------------------------------------------------------------------
extern "C" void kernel_launch(void* const* d_in, const int* in_sizes, int n_in,
                              void* d_out, int out_size, void* d_ws, size_t ws_size,
                              hipStream_t stream) {
  const float* deltas = (const float*)d_in[0];   // (8,64,64,36)
  const float* labels = (const float*)d_in[1];   // (8,64,64,9)
  const float* base   = (const float*)d_in[2];   // (9,4)
  float* out = (float*)d_out;                    // (8,300,4)

  float*    pre_boxes = (float*)d_ws;                                 // 256000 B
  unsigned* mask      = (unsigned*)((char*)d_ws + (1 << 18));          // 4.0 MB

  topk_decode_kernel<<<dim3(BATCH), dim3(1024), 0, stream>>>(labels, deltas, base, pre_boxes);
  dim3 g2(BATCH, (PRE_NMS * MASK_W + MASK_THREADS - 1) / MASK_THREADS);
  build_mask_kernel<<<g2, dim3(MASK_THREADS), 0, stream>>>(pre_boxes, mask);
  nms_scan_kernel<<<dim3(BATCH), dim3(64), 0, stream>>>(pre_boxes, mask, out);

  (void)in_sizes; (void)n_in; (void)out_size; (void)ws_size;
}